// GIN_3856880631940
// MI455X (gfx1250) — compile-verified
//
#include <hip/hip_runtime.h>
#include <hip/hip_bf16.h>

typedef __attribute__((ext_vector_type(2))) float v2f;
typedef __attribute__((ext_vector_type(8))) float v8f;

// ---------------------------------------------------------------------------
// Edge scatter: agg[dst] += x[src].  16 threads per edge, 4 floats each.
// x table (25.6MB) is L2-resident on MI455X (192MB L2) so gathers+atomics
// stay on-chip.
// ---------------------------------------------------------------------------
__global__ void scatter_kernel(const float* __restrict__ x,
                               const int* __restrict__ ei,
                               float* __restrict__ agg, int E) {
    int tid = blockIdx.x * blockDim.x + threadIdx.x;
    int e = tid >> 4;
    if (e >= E) return;
    int c = (tid & 15) * 4;
    int src = ei[e];
    int dst = ei[E + e];
    const float4 v = *(const float4*)(x + (size_t)src * 64 + c);
    float* p = agg + (size_t)dst * 64 + c;
    atomicAdd(p + 0, v.x);
    atomicAdd(p + 1, v.y);
    atomicAdd(p + 2, v.z);
    atomicAdd(p + 3, v.w);
}

// ---------------------------------------------------------------------------
// Generic fused WMMA GEMM:  out[16-row tile] = pre(A) @ W + bias
//   pre(A) = relu_if_sb( sb0[c]*((1+eps)*A + A2) + sb1[c] )
//   - epsp: device ptr to eps scalar (nullptr -> 1.0)
//   - A2:   optional second operand added elementwise (GIN aggregation)
//   - sb:   optional folded BN scale/bias [K]+[K]; when present also ReLU
// One wave per 16x16 output tile; blockDim = 32 * (ncols/16).
// A tile staged in LDS with stride K+4 (bank-conflict-free column reads).
// Uses V_WMMA_F32_16X16X4_F32 (exact fp32, matches reference numerics).
// ---------------------------------------------------------------------------
__global__ void gemm_tile_kernel(const float* __restrict__ A,
                                 const float* __restrict__ A2,
                                 const float* __restrict__ epsp,
                                 const float* __restrict__ sb,
                                 const float* __restrict__ W,
                                 const float* __restrict__ bias,
                                 float* __restrict__ out,
                                 int K, int ncols) {
    extern __shared__ float lds[];
    const int tid = threadIdx.x;
    const int rowBase = blockIdx.x * 16;
    const int st = K + 4;

    const float cA = epsp ? (1.0f + epsp[0]) : 1.0f;

    // Stage 16 x K tile of pre(A) into LDS (float4, 16B aligned).
    for (int idx = tid * 4; idx < 16 * K; idx += blockDim.x * 4) {
        const int row = idx / K;
        const int c = idx % K;
        float4 val = *(const float4*)(A + (size_t)(rowBase + row) * K + c);
        val.x *= cA; val.y *= cA; val.z *= cA; val.w *= cA;
        if (A2) {
            const float4 a2 = *(const float4*)(A2 + (size_t)(rowBase + row) * K + c);
            val.x += a2.x; val.y += a2.y; val.z += a2.z; val.w += a2.w;
        }
        if (sb) {
            val.x = fmaxf(sb[c + 0] * val.x + sb[K + c + 0], 0.0f);
            val.y = fmaxf(sb[c + 1] * val.y + sb[K + c + 1], 0.0f);
            val.z = fmaxf(sb[c + 2] * val.z + sb[K + c + 2], 0.0f);
            val.w = fmaxf(sb[c + 3] * val.w + sb[K + c + 3], 0.0f);
        }
        *(float4*)&lds[row * st + c] = val;
    }
    __syncthreads();

    // Per-wave 16x16 tile, K-loop of fp32 WMMA (16x16x4).
    const int lane = tid & 31;
    const int wave = tid >> 5;
    const int col = wave * 16 + (lane & 15);
    const int m = lane & 15;
    const int koff = (lane >> 4) * 2;   // A: lanes 0-15 hold K0,K1; 16-31 hold K2,K3

    v8f acc = {};
    for (int k0 = 0; k0 < K; k0 += 4) {
        const int kk = k0 + koff;
        v2f a, b;
        a.x = lds[m * st + kk];
        a.y = lds[m * st + kk + 1];
        b.x = W[(size_t)kk * ncols + col];
        b.y = W[(size_t)(kk + 1) * ncols + col];
        acc = __builtin_amdgcn_wmma_f32_16x16x4_f32(
            /*neg_a=*/false, a, /*neg_b=*/false, b,
            /*c_mod=*/(short)0, acc, /*reuse_a=*/false, /*reuse_b=*/false);
    }

    // D layout: VGPR i -> M=i (lanes 0-15) / M=8+i (lanes 16-31), N=lane&15.
    const float bv = bias[col];
    const int row0 = rowBase + (lane >> 4) * 8;
#pragma unroll
    for (int i = 0; i < 8; i++)
        out[(size_t)(row0 + i) * ncols + col] = acc[i] + bv;
}

// ---------------------------------------------------------------------------
// Column sums / sums-of-squares for training-mode BN.  sums[0..ncols) = sum,
// sums[ncols..2*ncols) = sumsq.  Must be zeroed before launch.
// ---------------------------------------------------------------------------
__global__ void colstats_kernel(const float* __restrict__ m, int nrows, int ncols,
                                float* __restrict__ sums) {
    __shared__ float s1[256];
    __shared__ float s2[256];
    const int tid = threadIdx.x;
    const int col = tid & (ncols - 1);
    const int sub = tid / ncols;
    const int tpc = blockDim.x / ncols;
    float a = 0.0f, b = 0.0f;
    for (int r = blockIdx.x * tpc + sub; r < nrows; r += gridDim.x * tpc) {
        const float v = m[(size_t)r * ncols + col];
        a += v;
        b += v * v;
    }
    s1[tid] = a;
    s2[tid] = b;
    __syncthreads();
    if (tid < ncols) {
        for (int j = 1; j < tpc; j++) {
            a += s1[tid + j * ncols];
            b += s2[tid + j * ncols];
        }
        atomicAdd(&sums[tid], a);
        atomicAdd(&sums[ncols + tid], b);
    }
}

// Fold BN into scale/bias:  bn(x) = sb[c]*x + sb[ncols+c]
__global__ void finalize_kernel(const float* __restrict__ sums,
                                const float* __restrict__ g,
                                const float* __restrict__ be,
                                int ncols, float invn, float* __restrict__ sb) {
    const int c = threadIdx.x;
    if (c >= ncols) return;
    const float mean = sums[c] * invn;
    const float var = fmaxf(sums[ncols + c] * invn - mean * mean, 0.0f);
    const float rstd = rsqrtf(var + 1e-5f);
    const float sc = g[c] * rstd;
    sb[c] = sc;
    sb[ncols + c] = be[c] - sc * mean;
}

// Elementwise BN + ReLU (outer batchnorm producing next-layer x)
__global__ void bn_relu_kernel(const float* __restrict__ v,
                               const float* __restrict__ sb,
                               float* __restrict__ out, int total, int ncols) {
    const int i = blockIdx.x * blockDim.x + threadIdx.x;
    if (i >= total) return;
    const int c = i & (ncols - 1);
    out[i] = fmaxf(sb[c] * v[i] + sb[ncols + c], 0.0f);
}

// Global add pool: gpool[batch[n]] += x[n]
__global__ void pool_kernel(const float* __restrict__ x,
                            const int* __restrict__ batch,
                            float* __restrict__ gpool, int total) {
    const int i = blockIdx.x * blockDim.x + threadIdx.x;
    if (i >= total) return;
    const int node = i >> 6;
    const int c = i & 63;
    atomicAdd(&gpool[(size_t)batch[node] * 64 + c], x[i]);
}

// ---------------------------------------------------------------------------
extern "C" void kernel_launch(void* const* d_in, const int* in_sizes, int n_in,
                              void* d_out, int out_size, void* d_ws, size_t ws_size,
                              hipStream_t stream) {
    const float* x_in = (const float*)d_in[0];
    const int* ei     = (const int*)d_in[1];
    const int* batch  = (const int*)d_in[2];
    const float* W1   = (const float*)d_in[4];
    const float* b1   = (const float*)d_in[5];
    const float* g1   = (const float*)d_in[6];
    const float* be1  = (const float*)d_in[7];
    const float* W2   = (const float*)d_in[8];
    const float* b2   = (const float*)d_in[9];
    const float* epsA = (const float*)d_in[10];
    const float* gc   = (const float*)d_in[11];
    const float* bc   = (const float*)d_in[12];
    const float* Wm1  = (const float*)d_in[13];
    const float* bm1  = (const float*)d_in[14];
    const float* gm   = (const float*)d_in[15];
    const float* bm   = (const float*)d_in[16];
    const float* Wm2  = (const float*)d_in[17];
    const float* bm2  = (const float*)d_in[18];

    const int N = in_sizes[0] / 64;   // 100000 (multiple of 16)
    const int E = in_sizes[1] / 2;    // 1600000
    const int L = in_sizes[10];       // 3
    const int Bg = out_size / 16;     // 1024 graphs (out is [B,16])

    // Workspace layout (floats). v aliases agg (dead after GEMM1).
    float* ws = (float*)d_ws;
    size_t o = 0;
    float* agg   = ws + o; o += (size_t)N * 64;    // also 'v'
    float* u     = ws + o; o += (size_t)N * 128;
    float* xA    = ws + o; o += (size_t)N * 64;
    float* xB    = ws + o; o += (size_t)N * 64;
    float* gpool = ws + o; o += (size_t)Bg * 64;
    float* ur    = ws + o; o += (size_t)Bg * 64;
    float* sums  = ws + o; o += 256;
    float* sb    = ws + o; o += 256;
    float* v = agg;

    const float invN = 1.0f / (float)N;
    const float invB = 1.0f / (float)Bg;

    for (int l = 0; l < L; l++) {
        const float* xcur = (l == 0) ? x_in : ((l & 1) ? xA : xB);
        float* xnext = (l & 1) ? xB : xA;

        // agg = segment_sum(x[src], dst)
        hipMemsetAsync(agg, 0, (size_t)N * 64 * sizeof(float), stream);
        scatter_kernel<<<(E * 16 + 255) / 256, 256, 0, stream>>>(xcur, ei, agg, E);

        // u = ((1+eps)*x + agg) @ W1 + b1      [N,128]
        gemm_tile_kernel<<<N / 16, 256, 16 * (64 + 4) * sizeof(float), stream>>>(
            xcur, agg, epsA + l, nullptr, W1 + (size_t)l * 64 * 128, b1 + (size_t)l * 128,
            u, 64, 128);

        // BN stats over u, fold into scale/bias
        hipMemsetAsync(sums, 0, 256 * sizeof(float), stream);
        colstats_kernel<<<512, 256, 0, stream>>>(u, N, 128, sums);
        finalize_kernel<<<1, 128, 0, stream>>>(sums, g1 + (size_t)l * 128,
                                               be1 + (size_t)l * 128, 128, invN, sb);

        // v = relu(bn(u)) @ W2 + b2            [N,64]
        gemm_tile_kernel<<<N / 16, 128, 16 * (128 + 4) * sizeof(float), stream>>>(
            u, nullptr, nullptr, sb, W2 + (size_t)l * 128 * 64, b2 + (size_t)l * 64,
            v, 128, 64);

        // outer BN + ReLU -> next x
        hipMemsetAsync(sums, 0, 256 * sizeof(float), stream);
        colstats_kernel<<<512, 256, 0, stream>>>(v, N, 64, sums);
        finalize_kernel<<<1, 64, 0, stream>>>(sums, gc + (size_t)l * 64,
                                              bc + (size_t)l * 64, 64, invN, sb);
        bn_relu_kernel<<<(N * 64 + 255) / 256, 256, 0, stream>>>(v, sb, xnext, N * 64, 64);
    }

    const float* xfin = ((L - 1) & 1) ? xB : xA;

    // global add pool
    hipMemsetAsync(gpool, 0, (size_t)Bg * 64 * sizeof(float), stream);
    pool_kernel<<<(N * 64 + 255) / 256, 256, 0, stream>>>(xfin, batch, gpool, N * 64);

    // readout: ur = gpool @ Wm1 + bm1
    gemm_tile_kernel<<<Bg / 16, 128, 16 * (64 + 4) * sizeof(float), stream>>>(
        gpool, nullptr, nullptr, nullptr, Wm1, bm1, ur, 64, 64);

    hipMemsetAsync(sums, 0, 256 * sizeof(float), stream);
    colstats_kernel<<<64, 256, 0, stream>>>(ur, Bg, 64, sums);
    finalize_kernel<<<1, 64, 0, stream>>>(sums, gm, bm, 64, invB, sb);

    // out = relu(bn(ur)) @ Wm2 + bm2           [B,16]
    gemm_tile_kernel<<<Bg / 16, 32, 16 * (64 + 4) * sizeof(float), stream>>>(
        ur, nullptr, nullptr, sb, Wm2, bm2, (float*)d_out, 64, 16);
}